// GCNLayer_49959059587405
// MI455X (gfx1250) — compile-verified
//
#include <hip/hip_runtime.h>

typedef __attribute__((ext_vector_type(2))) float v2f;
typedef __attribute__((ext_vector_type(8))) float v8f;

#define NN 8192
#define FF 256

// ---------------------------------------------------------------------------
// Kernel 1: dis[i] = rsqrt(sum_j A[i][j])   (one wave32 per row, float4 loads)
// ---------------------------------------------------------------------------
__global__ __launch_bounds__(256) void gcn_rowsum(const float* __restrict__ A,
                                                  float* __restrict__ dis) {
  const int wave = threadIdx.x >> 5;
  const int lane = threadIdx.x & 31;
  const int row  = blockIdx.x * 8 + wave;
  const float* a = A + (size_t)row * NN;
  float s = 0.0f;
  for (int k = lane * 4; k < NN; k += 32 * 4) {
    float4 v = *(const float4*)(a + k);
    s += (v.x + v.y) + (v.z + v.w);
  }
#pragma unroll
  for (int off = 16; off > 0; off >>= 1) s += __shfl_xor(s, off, 32);
  if (lane == 0) dis[row] = rsqrtf(s);
}

// ---------------------------------------------------------------------------
// Kernel 2: Hs[j][f] = dis[j] * H[j][f]
// ---------------------------------------------------------------------------
__global__ __launch_bounds__(256) void gcn_scaleH(const float* __restrict__ H,
                                                  const float* __restrict__ dis,
                                                  float* __restrict__ Hs) {
  unsigned i = blockIdx.x * 256u + threadIdx.x;   // covers NN*FF elements
  Hs[i] = H[i] * dis[i >> 8];                     // FF == 256
}

// ---------------------------------------------------------------------------
// Generic WMMA-f32 GEMM:  C[M,256] = op( A[M,K] @ B[K,256] )
//   SCALE: multiply output row i by dis[i]   (gemm1 epilogue)
//   RELU:  clamp output at 0                 (gemm2 epilogue)
// Wave tile: 32 rows x 32 cols = 2x2 WMMA tiles (4 v8f accumulators).
// Block: 8 waves = 8 column groups over the SAME 32 rows -> A fragments
// shared block-wide in WGP$; A streams from HBM exactly once.
// Per K-step(4): 2 b64 A-loads + 4 b32 B-loads feed 4 wmmas.
// ---------------------------------------------------------------------------
template <int K, bool SCALE, bool RELU>
__global__ __launch_bounds__(256) void gcn_gemm(const float* __restrict__ Am,
                                                const float* __restrict__ Bm,
                                                const float* __restrict__ dis,
                                                float* __restrict__ Cm) {
  const int wv   = threadIdx.x >> 5;     // column group 0..7
  const int lane = threadIdx.x & 31;
  const int half = lane >> 4;            // 0: K=k,k+1   1: K=k+2,k+3
  const int l    = lane & 15;
  const int row0 = blockIdx.x * 32;
  const int col0 = wv * 32;

  const float* __restrict__ Arow0 = Am + (size_t)(row0 + l) * K + 2 * half;
  const float* __restrict__ Arow1 = Arow0 + (size_t)16 * K;
  const float* __restrict__ Bp    = Bm + (size_t)(2 * half) * FF + col0 + l;

  v8f c00 = {}, c01 = {}, c10 = {}, c11 = {};

#pragma unroll 4
  for (int k = 0; k < K; k += 4) {
    v2f a0 = *(const v2f*)(Arow0 + k);          // rows row0   .. row0+15
    v2f a1 = *(const v2f*)(Arow1 + k);          // rows row0+16.. row0+31
    const float* bp = Bp + (size_t)k * FF;
    v2f b0; b0.x = bp[0];  b0.y = bp[FF];       // cols col0   .. col0+15
    v2f b1; b1.x = bp[16]; b1.y = bp[16 + FF];  // cols col0+16.. col0+31
    c00 = __builtin_amdgcn_wmma_f32_16x16x4_f32(false, a0, false, b0, (short)0, c00, false, false);
    c01 = __builtin_amdgcn_wmma_f32_16x16x4_f32(false, a0, false, b1, (short)0, c01, false, false);
    c10 = __builtin_amdgcn_wmma_f32_16x16x4_f32(false, a1, false, b0, (short)0, c10, false, false);
    c11 = __builtin_amdgcn_wmma_f32_16x16x4_f32(false, a1, false, b1, (short)0, c11, false, false);
  }

  // C/D layout: element r of v8f in lane -> row = r + 8*half, col = l.
#pragma unroll
  for (int u = 0; u < 2; ++u) {
    const v8f& ca = u ? c10 : c00;
    const v8f& cb = u ? c11 : c01;
#pragma unroll
    for (int r = 0; r < 8; ++r) {
      const int row = row0 + u * 16 + half * 8 + r;
      float va = ca[r];
      float vb = cb[r];
      if (SCALE) { const float d = dis[row]; va *= d; vb *= d; }
      if (RELU)  { va = fmaxf(va, 0.0f); vb = fmaxf(vb, 0.0f); }
      const size_t o = (size_t)row * FF + col0 + l;
      Cm[o]      = va;
      Cm[o + 16] = vb;
    }
  }
}

// ---------------------------------------------------------------------------
extern "C" void kernel_launch(void* const* d_in, const int* in_sizes, int n_in,
                              void* d_out, int out_size, void* d_ws, size_t ws_size,
                              hipStream_t stream) {
  const float* H = (const float*)d_in[0];   // [8192, 256]
  const float* A = (const float*)d_in[1];   // [8192, 8192]
  const float* W = (const float*)d_in[2];   // [256, 256]
  float* out = (float*)d_out;               // [8192, 256]

  float* ws  = (float*)d_ws;
  float* dis = ws;                          // 8192 floats
  float* Hs  = ws + NN;                     // 8192*256 floats
  float* M   = Hs + (size_t)NN * FF;        // 8192*256 floats

  gcn_rowsum<<<NN / 8, 256, 0, stream>>>(A, dis);
  gcn_scaleH<<<(NN * FF) / 256, 256, 0, stream>>>(H, dis, Hs);
  // M = diag(dis) * (A @ Hs)
  gcn_gemm<NN, true, false><<<NN / 32, 256, 0, stream>>>(A, Hs, dis, M);
  // out = relu(M @ W)
  gcn_gemm<FF, false, true><<<NN / 32, 256, 0, stream>>>(M, W, nullptr, out);
}